// AudioNetwork_75746043232861
// MI455X (gfx1250) — compile-verified
//
#include <hip/hip_runtime.h>
#include <math.h>

// Problem constants (from reference)
#define T_LEN    262144
#define BATCH    32
#define WSIZE    2048
#define STEP     1024
#define NC       1025
#define NFRAMES  256
#define NB       4
#define NPAD     2112                 // 33*64 : padded spectral column count (2*NC=2050 -> 2112)
#define M_TOTAL  (NFRAMES * BATCH)    // 8192 rows (frames x batch)
#define ACC_STRIDE (T_LEN + STEP)     // per-batch OLA accumulator row

// LDS tile stride: rows of 16 k-values padded to 20 floats.
// b64 fragment reads: banks 20*l mod 64 are 16 distinct multiples of 4 -> with the
// half-wave +2 column shift the 32 lanes x 2 dwords cover all 64 banks exactly once.
#define AS_STRIDE 20

// Workspace layout (in floats)
#define SZ_BFWD  ((size_t)NPAD * WSIZE)        // BfwdT: 2112 x 2048 (j-major, k contiguous)
#define SZ_BINV  ((size_t)WSIZE * NPAD)        // BinvT: 2048 x 2112 (n-major, k' contiguous)
#define SZ_SPEC  ((size_t)M_TOTAL * NPAD)      // 8192 x 2112 spectrum (Re | Im | zero-pad)
#define SZ_ACC   ((size_t)BATCH * ACC_STRIDE)  // overlap-add accumulator / stage signal
#define OFF_BFWD 0ull
#define OFF_BINV (OFF_BFWD + SZ_BFWD)
#define OFF_SPEC (OFF_BINV + SZ_BINV)
#define OFF_ACC  (OFF_SPEC + SZ_SPEC)
#define OFF_W    (OFF_ACC + SZ_ACC)

typedef float v2f __attribute__((ext_vector_type(2)));
typedef float v8f __attribute__((ext_vector_type(8)));

// ---------------------------------------------------------------------------
// Init: softmax of mixer (5 entries)
__global__ void init_weights_kernel(const float* __restrict__ mixer, float* __restrict__ w) {
    if (threadIdx.x == 0) {
        float m = mixer[0];
        for (int i = 1; i < NB + 1; ++i) m = fmaxf(m, mixer[i]);
        float s = 0.0f, e[NB + 1];
        for (int i = 0; i < NB + 1; ++i) { e[i] = __expf(mixer[i] - m); s += e[i]; }
        float inv = 1.0f / s;
        for (int i = 0; i < NB + 1; ++i) w[i] = e[i] * inv;
    }
}

// ---------------------------------------------------------------------------
// Init: TRANSPOSED DFT tables (reduction dim k contiguous in each row).
//  BfwdT[j][k], j=0..NPAD-1, k=0..WSIZE-1:
//     j<NC  -> cos(2*pi*j*k/W) ; j<2NC -> -sin(2*pi*(j-NC)*k/W) ; else 0
//  BinvT[n][k'], n=0..WSIZE-1, k'=0..NPAD-1:
//     k'<NC -> cos(2*pi*k'*n/W) ; k'<2NC -> +sin(2*pi*(k'-NC)*n/W) ; else 0
// Integer mod-2048 argument reduction keeps the tables exact to f32.
__global__ void init_tables_kernel(float* __restrict__ BfwdT, float* __restrict__ BinvT) {
    const float w0 = 6.283185307179586f / (float)WSIZE;
    const size_t total = SZ_BFWD; // == SZ_BINV
    for (size_t idx = (size_t)blockIdx.x * blockDim.x + threadIdx.x; idx < total;
         idx += (size_t)gridDim.x * blockDim.x) {
        int j = (int)(idx / WSIZE);
        int k = (int)(idx % WSIZE);
        float v = 0.0f;
        if (j < NC)          v =  cosf(w0 * (float)((j * k) & (WSIZE - 1)));
        else if (j < 2 * NC) v = -sinf(w0 * (float)(((j - NC) * k) & (WSIZE - 1)));
        BfwdT[idx] = v;

        int n  = (int)(idx / NPAD);
        int kp = (int)(idx % NPAD);
        float u = 0.0f;
        if (kp < NC)          u = cosf(w0 * (float)((kp * n) & (WSIZE - 1)));
        else if (kp < 2 * NC) u = sinf(w0 * (float)(((kp - NC) * n) & (WSIZE - 1)));
        BinvT[idx] = u;
    }
}

// ---------------------------------------------------------------------------
// d_out = w[0] * x
__global__ void init_out_kernel(const float* __restrict__ x, const float* __restrict__ w,
                                float* __restrict__ out) {
    size_t idx = (size_t)blockIdx.x * blockDim.x + threadIdx.x;
    if (idx < (size_t)BATCH * T_LEN) out[idx] = w[0] * x[idx];
}

__global__ void zero_acc_kernel(float* __restrict__ acc) {
    size_t idx = (size_t)blockIdx.x * blockDim.x + threadIdx.x;
    if (idx < SZ_ACC) acc[idx] = 0.0f;
}

// ---------------------------------------------------------------------------
// Forward rDFT as GEMM: spec[m][j] = sum_k frame(m,k) * BfwdT[j][k]
// m = f*32 + b ; frame(m,k) = cur[b][f*STEP + k] (zero past T_LEN).
// Macro tile 128x64, 8 waves (4x2), wave tile 32x32 (4 WMMA accumulators).
// Double-buffered LDS ping-pong (1 barrier per K-step) + register prefetch.
__launch_bounds__(256)
__global__ void fwd_dft_kernel(const float* __restrict__ cur, size_t curStride,
                               const float* __restrict__ BfwdT, float* __restrict__ spec) {
    __shared__ float As0[128 * AS_STRIDE], As1[128 * AS_STRIDE];
    __shared__ float Bt0[64 * AS_STRIDE],  Bt1[64 * AS_STRIDE];

    const int tid  = threadIdx.x;
    const int wave = tid >> 5;
    const int lane = tid & 31;
    const int mTile = blockIdx.y;
    const int nTile = blockIdx.x;
    const int wm = wave >> 1;       // 0..3  (M sub-tile, 32 rows)
    const int wn = wave & 1;        // 0..1  (N sub-tile, 32 cols)
    const int lm = lane & 15;
    const int kb = (lane >> 4) * 2; // K half-split per ISA 7.12.2

    // Per-thread tile-load mapping: row r = tid>>2, 4 k's at kq = (tid&3)*4.
    const int r0  = tid >> 2;            // A rows 0..63 and B rows 0..63
    const int r1  = r0 + 64;             // A rows 64..127
    const int kq  = (tid & 3) * 4;

    const int am0 = mTile * 128 + r0, af0 = am0 >> 5, ab0 = am0 & 31;
    const int am1 = mTile * 128 + r1, af1 = am1 >> 5, ab1 = am1 & 31;
    const float* bRow = BfwdT + (size_t)(nTile * 64 + r0) * WSIZE;

    v8f c00 = {}, c01 = {}, c10 = {}, c11 = {};
    float4 ra0, ra1, rb;

    auto loadT = [&](int kbase) {
        int s0 = af0 * STEP + kbase + kq;
        int s1 = af1 * STEP + kbase + kq;
        ra0 = *(const float4*)(cur + (size_t)ab0 * curStride + (s0 < T_LEN ? s0 : 0));
        ra1 = *(const float4*)(cur + (size_t)ab1 * curStride + (s1 < T_LEN ? s1 : 0));
        ra0.x = (s0 < T_LEN) ? ra0.x : 0.0f; ra0.y = (s0 < T_LEN) ? ra0.y : 0.0f;
        ra0.z = (s0 < T_LEN) ? ra0.z : 0.0f; ra0.w = (s0 < T_LEN) ? ra0.w : 0.0f;
        ra1.x = (s1 < T_LEN) ? ra1.x : 0.0f; ra1.y = (s1 < T_LEN) ? ra1.y : 0.0f;
        ra1.z = (s1 < T_LEN) ? ra1.z : 0.0f; ra1.w = (s1 < T_LEN) ? ra1.w : 0.0f;
        rb = *(const float4*)(bRow + kbase + kq);
    };
    auto storeT = [&](float* Asb, float* Btb) {
        *(float4*)&Asb[r0 * AS_STRIDE + kq] = ra0;
        *(float4*)&Asb[r1 * AS_STRIDE + kq] = ra1;
        *(float4*)&Btb[r0 * AS_STRIDE + kq] = rb;
    };
    auto compute = [&](const float* Asb, const float* Btb) {
        #pragma unroll
        for (int kk = 0; kk < 4; ++kk) {
            v2f a0 = *(const v2f*)&Asb[(wm * 32 + lm)      * AS_STRIDE + kk * 4 + kb];
            v2f a1 = *(const v2f*)&Asb[(wm * 32 + 16 + lm) * AS_STRIDE + kk * 4 + kb];
            v2f b0 = *(const v2f*)&Btb[(wn * 32 + lm)      * AS_STRIDE + kk * 4 + kb];
            v2f b1 = *(const v2f*)&Btb[(wn * 32 + 16 + lm) * AS_STRIDE + kk * 4 + kb];
            c00 = __builtin_amdgcn_wmma_f32_16x16x4_f32(false, a0, false, b0, (short)0, c00, false, false);
            c01 = __builtin_amdgcn_wmma_f32_16x16x4_f32(false, a0, false, b1, (short)0, c01, false, false);
            c10 = __builtin_amdgcn_wmma_f32_16x16x4_f32(false, a1, false, b0, (short)0, c10, false, false);
            c11 = __builtin_amdgcn_wmma_f32_16x16x4_f32(false, a1, false, b1, (short)0, c11, false, false);
        }
    };

    loadT(0);
    storeT(As0, Bt0);
    __syncthreads();
    for (int k0 = 0; k0 < WSIZE; k0 += 32) {
        // phase 0: compute buf0, prefetch k0+16 (always in range), store buf1
        loadT(k0 + 16);
        compute(As0, Bt0);
        storeT(As1, Bt1);
        __syncthreads();
        // phase 1: compute buf1, prefetch k0+32 if any, store buf0
        const bool more = (k0 + 32) < WSIZE;
        if (more) loadT(k0 + 32);
        compute(As1, Bt1);
        if (more) storeT(As0, Bt0);
        __syncthreads();
    }

    const int half = lane >> 4;
    const int mB = mTile * 128 + wm * 32 + half * 8;
    const int nB = nTile * 64 + wn * 32 + lm;
    #pragma unroll
    for (int v = 0; v < 8; ++v) {
        spec[(size_t)(mB + v)      * NPAD + nB]      = c00[v];
        spec[(size_t)(mB + v)      * NPAD + nB + 16] = c01[v];
        spec[(size_t)(mB + 16 + v) * NPAD + nB]      = c10[v];
        spec[(size_t)(mB + 16 + v) * NPAD + nB + 16] = c11[v];
    }
}

// ---------------------------------------------------------------------------
// Per-bin IIR across frames, in place, folding irfft weights:
//   carry = (spec + carry) * t ;  spec.Re <- ck*carry.Re ; spec.Im <- -ck*carry.Im
__global__ void scan_kernel(float* __restrict__ spec, const float* __restrict__ transfers, int blk) {
    int k = blockIdx.x * blockDim.x + threadIdx.x;
    int b = blockIdx.y;
    if (k >= NC) return;
    float tr = transfers[blk * NC + k];
    float ck = ((k == 0) || (k == NC - 1)) ? (1.0f / WSIZE) : (2.0f / WSIZE);
    float cre = 0.0f, cim = 0.0f;
    for (int f = 0; f < NFRAMES; ++f) {
        size_t base = (size_t)(f * BATCH + b) * NPAD;
        float re = spec[base + k];
        float im = spec[base + NC + k];
        cre = (re + cre) * tr;
        cim = (im + cim) * tr;
        spec[base + k]      =  ck * cre;
        spec[base + NC + k] = -ck * cim;
    }
}

// ---------------------------------------------------------------------------
// Inverse rDFT as GEMM with fused Hann + overlap-add epilogue:
//   win(m,n) = sum_k spec[m][k] * BinvT[n][k] ; acc[b][f*STEP+n] += win * hann(n)
__launch_bounds__(256)
__global__ void inv_dft_kernel(const float* __restrict__ spec, const float* __restrict__ BinvT,
                               float* __restrict__ acc) {
    __shared__ float As0[128 * AS_STRIDE], As1[128 * AS_STRIDE];
    __shared__ float Bt0[64 * AS_STRIDE],  Bt1[64 * AS_STRIDE];

    const int tid  = threadIdx.x;
    const int wave = tid >> 5;
    const int lane = tid & 31;
    const int mTile = blockIdx.y;
    const int nTile = blockIdx.x;
    const int wm = wave >> 1;
    const int wn = wave & 1;
    const int lm = lane & 15;
    const int kb = (lane >> 4) * 2;

    const int r0 = tid >> 2;
    const int r1 = r0 + 64;
    const int kq = (tid & 3) * 4;

    const float* aRow0 = spec + (size_t)(mTile * 128 + r0) * NPAD;
    const float* aRow1 = spec + (size_t)(mTile * 128 + r1) * NPAD;
    const float* bRow  = BinvT + (size_t)(nTile * 64 + r0) * NPAD;

    v8f c00 = {}, c01 = {}, c10 = {}, c11 = {};
    float4 ra0, ra1, rb;

    auto loadT = [&](int kbase) {
        ra0 = *(const float4*)(aRow0 + kbase + kq);
        ra1 = *(const float4*)(aRow1 + kbase + kq);
        rb  = *(const float4*)(bRow  + kbase + kq);
    };
    auto storeT = [&](float* Asb, float* Btb) {
        *(float4*)&Asb[r0 * AS_STRIDE + kq] = ra0;
        *(float4*)&Asb[r1 * AS_STRIDE + kq] = ra1;
        *(float4*)&Btb[r0 * AS_STRIDE + kq] = rb;
    };
    auto compute = [&](const float* Asb, const float* Btb) {
        #pragma unroll
        for (int kk = 0; kk < 4; ++kk) {
            v2f a0 = *(const v2f*)&Asb[(wm * 32 + lm)      * AS_STRIDE + kk * 4 + kb];
            v2f a1 = *(const v2f*)&Asb[(wm * 32 + 16 + lm) * AS_STRIDE + kk * 4 + kb];
            v2f b0 = *(const v2f*)&Btb[(wn * 32 + lm)      * AS_STRIDE + kk * 4 + kb];
            v2f b1 = *(const v2f*)&Btb[(wn * 32 + 16 + lm) * AS_STRIDE + kk * 4 + kb];
            c00 = __builtin_amdgcn_wmma_f32_16x16x4_f32(false, a0, false, b0, (short)0, c00, false, false);
            c01 = __builtin_amdgcn_wmma_f32_16x16x4_f32(false, a0, false, b1, (short)0, c01, false, false);
            c10 = __builtin_amdgcn_wmma_f32_16x16x4_f32(false, a1, false, b0, (short)0, c10, false, false);
            c11 = __builtin_amdgcn_wmma_f32_16x16x4_f32(false, a1, false, b1, (short)0, c11, false, false);
        }
    };

    loadT(0);
    storeT(As0, Bt0);
    __syncthreads();
    for (int k0 = 0; k0 < NPAD; k0 += 32) {
        loadT(k0 + 16);                       // k0+16 <= NPAD-16 always
        compute(As0, Bt0);
        storeT(As1, Bt1);
        __syncthreads();
        const bool more = (k0 + 32) < NPAD;
        if (more) loadT(k0 + 32);
        compute(As1, Bt1);
        if (more) storeT(As0, Bt0);
        __syncthreads();
    }

    const float w0 = 6.283185307179586f / (float)WSIZE;
    const int half = lane >> 4;
    const int mB = mTile * 128 + wm * 32 + half * 8;
    const int n0 = nTile * 64 + wn * 32 + lm;
    const int n1 = n0 + 16;
    const float h0 = 0.5f - 0.5f * cosf(w0 * (float)n0);
    const float h1 = 0.5f - 0.5f * cosf(w0 * (float)n1);
    #pragma unroll
    for (int v = 0; v < 8; ++v) {
        int m0 = mB + v;
        int m1 = mB + 16 + v;
        int f0 = m0 >> 5, b0 = m0 & 31;
        int f1 = m1 >> 5, b1 = m1 & 31;
        atomicAdd(&acc[(size_t)b0 * ACC_STRIDE + f0 * STEP + n0], c00[v] * h0);
        atomicAdd(&acc[(size_t)b0 * ACC_STRIDE + f0 * STEP + n1], c01[v] * h1);
        atomicAdd(&acc[(size_t)b1 * ACC_STRIDE + f1 * STEP + n0], c10[v] * h0);
        atomicAdd(&acc[(size_t)b1 * ACC_STRIDE + f1 * STEP + n1], c11[v] * h1);
    }
}

// ---------------------------------------------------------------------------
// s = tanh(acc * gain); acc <- s (becomes next block's input); out += w[blk+1]*s
__global__ void finalize_kernel(float* __restrict__ acc, const float* __restrict__ gains,
                                const float* __restrict__ w, float* __restrict__ out, int blk) {
    size_t idx = (size_t)blockIdx.x * blockDim.x + threadIdx.x;
    if (idx >= (size_t)BATCH * T_LEN) return;
    int b = (int)(idx >> 18);           // T_LEN = 2^18
    size_t t = idx & (size_t)(T_LEN - 1);
    float s = tanhf(acc[(size_t)b * ACC_STRIDE + t] * gains[blk]);
    acc[(size_t)b * ACC_STRIDE + t] = s;
    out[idx] += w[blk + 1] * s;
}

// ---------------------------------------------------------------------------
extern "C" void kernel_launch(void* const* d_in, const int* in_sizes, int n_in,
                              void* d_out, int out_size, void* d_ws, size_t ws_size,
                              hipStream_t stream) {
    const float* x         = (const float*)d_in[0];
    const float* transfers = (const float*)d_in[1];
    const float* gains     = (const float*)d_in[2];
    const float* mixer     = (const float*)d_in[3];
    float* out = (float*)d_out;
    float* ws  = (float*)d_ws;

    float* BfwdT = ws + OFF_BFWD;
    float* BinvT = ws + OFF_BINV;
    float* spec  = ws + OFF_SPEC;
    float* acc   = ws + OFF_ACC;
    float* wmix  = ws + OFF_W;

    init_weights_kernel<<<1, 32, 0, stream>>>(mixer, wmix);
    init_tables_kernel<<<4096, 256, 0, stream>>>(BfwdT, BinvT);
    init_out_kernel<<<(int)(((size_t)BATCH * T_LEN + 255) / 256), 256, 0, stream>>>(x, wmix, out);

    for (int blk = 0; blk < NB; ++blk) {
        const float* cur  = (blk == 0) ? x : acc;
        size_t curStride  = (blk == 0) ? (size_t)T_LEN : (size_t)ACC_STRIDE;
        fwd_dft_kernel<<<dim3(NPAD / 64, M_TOTAL / 128), 256, 0, stream>>>(cur, curStride, BfwdT, spec);
        scan_kernel<<<dim3((NC + 255) / 256, BATCH), 256, 0, stream>>>(spec, transfers, blk);
        zero_acc_kernel<<<(int)((SZ_ACC + 255) / 256), 256, 0, stream>>>(acc);
        inv_dft_kernel<<<dim3(WSIZE / 64, M_TOTAL / 128), 256, 0, stream>>>(spec, BinvT, acc);
        finalize_kernel<<<(int)(((size_t)BATCH * T_LEN + 255) / 256), 256, 0, stream>>>(acc, gains, wmix, out, blk);
    }
}